// EnergyMapping_63891933495828
// MI455X (gfx1250) — compile-verified
//
#include <hip/hip_runtime.h>
#include <hip/hip_bf16.h>

typedef float v2f __attribute__((ext_vector_type(2)));
typedef float v8f __attribute__((ext_vector_type(8)));

#define FDIM 128
#define HDIM 64
#define NB 16
#define EDGES_PER_BATCH (256 * 64)              // 16384
#define NEDGES (NB * EDGES_PER_BATCH)           // 262144
#define NTILES (NEDGES / 16)                    // 16384 tiles of 16 edges
#define TILES_PER_BATCH (EDGES_PER_BATCH / 16)  // 1024

__global__ void energy_init_kernel(float* __restrict__ out,
                                   const float* __restrict__ b2) {
    int i = threadIdx.x;
    if (i < NB) out[i] = (float)EDGES_PER_BATCH * b2[0];
}

__global__ __launch_bounds__(256)
void energy_mlp_kernel(const float* __restrict__ X,   // (NEDGES, 128)
                       const float* __restrict__ W1,  // (128, 64) row-major
                       const float* __restrict__ b1,  // (64)
                       const float* __restrict__ W2,  // (64)
                       float* __restrict__ out) {     // (16)
    // Pair-packed W1: w1p[pk*64 + n] = (W1[2pk][n], W1[2pk+1][n]).
    // A B-fragment for K-rows (k0, k0+1) is then ONE aligned ds_load_b64.
    __shared__ v2f w1p[(FDIM / 2) * HDIM]; // 4096 * 8B = 32 KB

    const int tid = threadIdx.x;

    #pragma unroll
    for (int i = 0; i < (FDIM / 2) * HDIM / 256; ++i) {
        int idx = tid + i * 256;      // 0..4095
        int pk  = idx >> 6;           // K-pair index (0..63)
        int n   = idx & 63;           // H column
        v2f p;
        p.x = W1[(2 * pk    ) * HDIM + n];
        p.y = W1[(2 * pk + 1) * HDIM + n];
        w1p[idx] = p;
    }
    __syncthreads();

    const int lane   = tid & 31;
    const int wave   = tid >> 5;
    const int col    = lane & 15;   // N (and M) index within 16-wide tile
    const int pkbase = lane >> 4;   // half-wave selects K-pair 0 or 1 of each quad

    // Per-lane bias / output-weight values for the 4 column tiles of H=64.
    float b1v[4], w2v[4];
    #pragma unroll
    for (int c = 0; c < 4; ++c) {
        b1v[c] = b1[c * 16 + col];
        w2v[c] = W2[c * 16 + col];
    }

    const int nwaves = gridDim.x * 8;
    const int gwave  = blockIdx.x * 8 + wave;
    const float LOG2 = 0.6931471805599453f;

    for (int t = gwave; t < NTILES; t += nwaves) {
        // A-fragment rows: 16 consecutive edges, each lane owns row M = col.
        const float* __restrict__ xrow = X + (size_t)(t * 16 + col) * FDIM;

        v8f acc[4] = {}; // 4 tiles of 16x16 f32 accumulators (32 VGPRs)

        #pragma unroll 4
        for (int kk = 0; kk < FDIM / 4; ++kk) {
            const int pk = 2 * kk + pkbase;   // K-pair this half-wave owns
            // A: lane holds X[row][2pk], X[row][2pk+1] (ISA 32-bit A layout)
            v2f a = *(const v2f*)(xrow + 2 * pk);
            const v2f* __restrict__ brow = w1p + pk * HDIM + col;
            #pragma unroll
            for (int c = 0; c < 4; ++c) {
                v2f b = brow[c * 16]; // (W1[k0][n], W1[k0+1][n]) in one b64
                acc[c] = __builtin_amdgcn_wmma_f32_16x16x4_f32(
                    /*neg_a=*/false, a, /*neg_b=*/false, b,
                    /*c_mod=*/(short)0, acc[c],
                    /*reuse_a=*/false, /*reuse_b=*/false);
            }
        }

        // Epilogue: h = softplus(acc + b1) - log2 ; partial += h * w2
        float p = 0.0f;
        #pragma unroll
        for (int c = 0; c < 4; ++c) {
            #pragma unroll
            for (int r = 0; r < 8; ++r) {
                float x  = acc[c][r] + b1v[c];
                float sp = fmaxf(x, 0.0f) + log1pf(__expf(-fabsf(x))) - LOG2;
                p = fmaf(sp, w2v[c], p);
            }
        }

        // Wave32 tree reduction: every lane/r covers a distinct (M,N) element.
        #pragma unroll
        for (int off = 16; off > 0; off >>= 1)
            p += __shfl_xor(p, off, 32);

        if (lane == 0) {
            const int b = t / TILES_PER_BATCH; // batch this 16-edge tile belongs to
            atomicAdd(&out[b], p);
        }
    }
}

extern "C" void kernel_launch(void* const* d_in, const int* in_sizes, int n_in,
                              void* d_out, int out_size, void* d_ws, size_t ws_size,
                              hipStream_t stream) {
    const float* X  = (const float*)d_in[0]; // edge_embedding (16,256,64,128)
    const float* W1 = (const float*)d_in[1]; // (128,64)
    const float* b1 = (const float*)d_in[2]; // (64,)
    const float* W2 = (const float*)d_in[3]; // (64,1)
    const float* b2 = (const float*)d_in[4]; // (1,)
    float* out = (float*)d_out;              // (16,1)

    energy_init_kernel<<<1, 32, 0, stream>>>(out, b2);
    energy_mlp_kernel<<<512, 256, 0, stream>>>(X, W1, b1, W2, out);
}